// Mul_Non_local_81157702025716
// MI455X (gfx1250) — compile-verified
//
#include <hip/hip_runtime.h>
#include <hip/hip_bf16.h>

typedef __attribute__((ext_vector_type(2))) float v2f;
typedef __attribute__((ext_vector_type(8))) float v8f;

#define NPIX  4096     // H*W
#define BATCH 4
#define BN_EPS 1e-5f

// Async gather from global into LDS (per-lane scatter, ASYNCcnt-tracked).
// Generic shared pointers carry the LDS byte offset in their low 32 bits.
__device__ __forceinline__ void async_g2l_b32(const float* gsrc, const float* ldst) {
  unsigned int loff = (unsigned int)(unsigned long long)ldst;
  asm volatile("global_load_async_to_lds_b32 %0, %1, off"
               :: "v"(loff), "v"(gsrc) : "memory");
}
__device__ __forceinline__ void wait_async_zero() {
  asm volatile("s_wait_asynccnt 0" ::: "memory");
}

// ---------------------------------------------------------------------------
// Kernel 1: fold 1x1 + 3x3 + 5x5 weights (per projection) into one effective
// 5x5 kernel, laid out channel-fastest: weff[(tap*4 + proj)*256 + ch]
// (proj 3 zero-padded) so B fragments are single b64 loads. Sum biases.
// ---------------------------------------------------------------------------
struct PrepArgs {
  const float* w1[3]; const float* w3[3]; const float* w5[3];
  const float* b1[3]; const float* b3[3]; const float* b5[3];
};

__global__ void prep_weights_kernel(PrepArgs pa, float* __restrict__ weff,
                                    float* __restrict__ bsum) {
  const int idx = blockIdx.x * blockDim.x + threadIdx.x;   // 25600 = 25*4*256
  const int ch  = idx & 255;
  const int p   = (idx >> 8) & 3;
  const int tap = idx >> 10;                               // 0..24
  float v = 0.f;
  if (p < 3) {
    const int ty = tap / 5, tx = tap % 5;
    v = pa.w5[p][ch * 25 + tap];
    if (ty >= 1 && ty <= 3 && tx >= 1 && tx <= 3)
      v += pa.w3[p][ch * 9 + (ty - 1) * 3 + (tx - 1)];
    if (tap == 12) v += pa.w1[p][ch];
  }
  weff[idx] = v;
  if (idx < 3) bsum[idx] = pa.b1[idx][0] + pa.b3[idx][0] + pa.b5[idx][0];
}

// ---------------------------------------------------------------------------
// Kernel 2: three 1-channel projections as a WMMA GEMM, LDS-staged.
// Block = 256 thr (8 waves) covers one batch b and a row pair (h0, h0+1):
// 8 tiles of 16 pixels. Per 32-channel chunk, the 6-row input slab is pulled
// into LDS with global_load_async_to_lds_b32 in a channel-pair-interleaved
// layout: slab[((cc>>1)*384 + rr*64 + col)*2 + (cc&1)], so an A fragment
// (K, K+1 channels at one pixel) is one ds_load_b64.
// D[16 pix x 16 cols]: cols 0..2 = g/theta/phi; A per ISA 16x4 f32 layout
// (lanes 0-15: K={0,1}; lanes 16-31: K={2,3}), B mirrored by column.
// ---------------------------------------------------------------------------
__global__ void __launch_bounds__(256) conv_wmma_kernel(
    const float* __restrict__ x, const float* __restrict__ weff,
    const float* __restrict__ bsum, float* __restrict__ maps) {
  __shared__ float slab[32 * 6 * 64];            // 48 KB

  const int lane = threadIdx.x & 31;
  const int wave = threadIdx.x >> 5;
  const int b  = blockIdx.x >> 5;                // 4 batches x 32 row-pairs
  const int h0 = (blockIdx.x & 31) << 1;
  const int rowInPair = wave >> 2;               // wave -> tile
  const int w0 = (wave & 3) << 4;
  const int h  = h0 + rowInPair;
  const int l16  = lane & 15;
  const int koff = (lane >> 4) << 1;             // 0 for lanes 0-15, 2 for 16-31
  const int wpix = w0 + l16;
  const bool wvalid = (l16 < 4);

  const float* xb = x + (size_t)b * (256 * NPIX);
  v8f acc = {};

  for (int c0 = 0; c0 < 256; c0 += 32) {
    // ---- stage 32ch x 6rows x 64cols slab into LDS (async, transposing) ----
    for (int i = 0; i < 48; ++i) {
      const int e   = threadIdx.x + (i << 8);    // 0..12287 = cc*384 + rr*64 + col
      const int col = e & 63;
      const int t   = e >> 6;
      const int rr  = t % 6;
      const int cc  = t / 6;
      const int hh  = h0 - 2 + rr;
      if (hh >= 0 && hh < 64) {
        const float* src = xb + (size_t)(c0 + cc) * NPIX + hh * 64 + col;
        const float* dst = &slab[(((cc >> 1) * 384 + rr * 64 + col) << 1) + (cc & 1)];
        async_g2l_b32(src, dst);
      }
    }
    wait_async_zero();
    __syncthreads();

    // ---- 25 taps x 8 K-steps of WMMA over this chunk ----
    for (int tap = 0; tap < 25; ++tap) {
      const int dy = tap / 5 - 2, dx = tap % 5 - 2;
      const int hh = h + dy, ww = wpix + dx;
      const bool valid = (hh >= 0) && (hh < 64) && (ww >= 0) && (ww < 64);
      const int rIdx  = rowInPair + dy + 2;                  // row inside slab
      const int abase = rIdx * 64 + ww;                      // v2f index base
      const v2f* ap = (const v2f*)slab;
      const float* wp = weff + (tap * 4 + l16) * 256 + c0 + koff;
#pragma unroll
      for (int cs = 0; cs < 32; cs += 4) {
        v2f av = {}, bv = {};
        if (valid)  av = ap[(((cs + koff) >> 1) * 384) + abase]; // ds_load_b64
        if (wvalid) bv = *(const v2f*)(wp + cs);                 // global b64
        acc = __builtin_amdgcn_wmma_f32_16x16x4_f32(false, av, false, bv,
                                                    (short)0, acc, false, false);
      }
    }
    __syncthreads();
  }

  // D layout: VGPR r, lanes 0-15 -> (M=r, N=lane); lanes 16-31 -> (M=r+8, N=lane-16)
  if (l16 < 3) {
    const float bias = bsum[l16];
    float* mp = maps + l16 * (BATCH * NPIX) + b * NPIX + h * 64 + w0 + ((lane >> 4) << 3);
#pragma unroll
    for (int r = 0; r < 8; ++r) mp[r] = acc[r] + bias;
  }
}

// ---------------------------------------------------------------------------
// Kernel 3: per-batch reductions. sred = [ s[4]=(phi.g)/N, t1[4]=sum(theta),
// t2[4]=sum(theta^2) ].
// ---------------------------------------------------------------------------
__global__ void reduce_kernel(const float* __restrict__ maps, float* __restrict__ sred) {
  const int b = blockIdx.x, tid = threadIdx.x;              // 4 blocks x 256
  const float* g = maps + b * NPIX;
  const float* t = maps + BATCH * NPIX + b * NPIX;
  const float* p = maps + 2 * BATCH * NPIX + b * NPIX;
  float s1 = 0.f, a1 = 0.f, a2 = 0.f;
  for (int i = tid; i < NPIX; i += 256) {
    const float gv = g[i], tv = t[i], pv = p[i];
    s1 += pv * gv; a1 += tv; a2 += tv * tv;
  }
  __shared__ float sh0[256], sh1[256], sh2[256];
  sh0[tid] = s1; sh1[tid] = a1; sh2[tid] = a2;
  __syncthreads();
  for (int st = 128; st > 0; st >>= 1) {
    if (tid < st) { sh0[tid] += sh0[tid + st]; sh1[tid] += sh1[tid + st]; sh2[tid] += sh2[tid + st]; }
    __syncthreads();
  }
  if (tid == 0) {
    sred[b]     = sh0[0] / (float)NPIX;  // s_b
    sred[4 + b] = sh1[0];
    sred[8 + b] = sh2[0];
  }
}

// ---------------------------------------------------------------------------
// Kernel 4: BN stats of W_y fold to stats of y; per-channel affine coef:
// a_c = gamma*wW*rsqrt(wW^2*var+eps); d_c = beta - a_c*mean. (bW cancels)
// ---------------------------------------------------------------------------
__global__ void coeff_kernel(const float* __restrict__ sred, const float* __restrict__ wW,
                             const float* __restrict__ gamma, const float* __restrict__ beta,
                             float* __restrict__ coef) {
  __shared__ float mv[2];
  if (threadIdx.x == 0) {
    float sy = 0.f, sy2 = 0.f;
    for (int b = 0; b < BATCH; ++b) {
      const float sb = sred[b];
      sy  += sb * sred[4 + b];
      sy2 += sb * sb * sred[8 + b];
    }
    const float inv = 1.0f / (float)(BATCH * NPIX);
    const float mean = sy * inv;
    mv[0] = mean;
    mv[1] = sy2 * inv - mean * mean;
  }
  __syncthreads();
  const int c = threadIdx.x;
  const float w = wW[c];
  const float a = gamma[c] * w * rsqrtf(w * w * mv[1] + BN_EPS);
  coef[c]       = a;
  coef[256 + c] = beta[c] - a * mv[0];
}

// ---------------------------------------------------------------------------
// Kernel 5: out[b,c,n] = a_c * (theta[b,n]*s_b) + d_c + x[b,c,n], float4-wide.
// ---------------------------------------------------------------------------
__global__ void finalize_kernel(const float* __restrict__ x, const float* __restrict__ tmap,
                                const float* __restrict__ sred, const float* __restrict__ coef,
                                float* __restrict__ out) {
  const int i  = blockIdx.x * blockDim.x + threadIdx.x;  // 0..1048575 float4s
  const int n4 = i & 1023;
  const int c  = (i >> 10) & 255;
  const int b  = i >> 18;
  const float4 xv = ((const float4*)x)[i];
  const float4 tv = ((const float4*)tmap)[b * 1024 + n4];
  const float sb = sred[b];
  const float a = coef[c], d = coef[256 + c];
  float4 o;
  o.x = fmaf(a, tv.x * sb, d) + xv.x;
  o.y = fmaf(a, tv.y * sb, d) + xv.y;
  o.z = fmaf(a, tv.z * sb, d) + xv.z;
  o.w = fmaf(a, tv.w * sb, d) + xv.w;
  ((float4*)out)[i] = o;
}

// ---------------------------------------------------------------------------
extern "C" void kernel_launch(void* const* d_in, const int* in_sizes, int n_in,
                              void* d_out, int out_size, void* d_ws, size_t ws_size,
                              hipStream_t stream) {
  const float* x = (const float*)d_in[0];
  PrepArgs pa;
  for (int grp = 0; grp < 3; ++grp) {           // g, t, p
    const int base = 1 + grp * 6;               // w1,b1,w2,b2,w3,b3
    pa.w1[grp] = (const float*)d_in[base + 0];
    pa.b1[grp] = (const float*)d_in[base + 1];
    pa.w3[grp] = (const float*)d_in[base + 2];
    pa.b3[grp] = (const float*)d_in[base + 3];
    pa.w5[grp] = (const float*)d_in[base + 4];
    pa.b5[grp] = (const float*)d_in[base + 5];
  }
  const float* wW    = (const float*)d_in[19];
  // d_in[20] = bW: cancels inside BatchNorm, unused.
  const float* gamma = (const float*)d_in[21];
  const float* beta  = (const float*)d_in[22];

  float* ws   = (float*)d_ws;
  float* weff = ws;                 // 25600 floats
  float* maps = ws + 25600;         // 3 * 16384 (g, theta, phi)
  float* bsum = maps + 3 * 16384;   // 4
  float* sred = bsum + 4;           // 12  (s, t1, t2)
  float* coef = sred + 16;          // 512 (a_c, d_c)   -> total ~295 KB

  prep_weights_kernel<<<100, 256, 0, stream>>>(pa, weff, bsum);
  conv_wmma_kernel<<<128, 256, 0, stream>>>(x, weff, bsum, maps);
  reduce_kernel<<<BATCH, 256, 0, stream>>>(maps, sred);
  coeff_kernel<<<1, 256, 0, stream>>>(sred, wW, gamma, beta, coef);
  finalize_kernel<<<4096, 256, 0, stream>>>(x, maps + BATCH * NPIX, sred, coef,
                                            (float*)d_out);
}